// JSMLP_25125558682019
// MI455X (gfx1250) — compile-verified
//
#include <hip/hip_runtime.h>
#include <hip/hip_bf16.h>

// ---------------------------------------------------------------------------
// Mixture-of-experts 3-layer MLP for gfx1250 (MI455X), f32 WMMA 16x16x4 path.
//   N=16384 rows, IN=64, H1=H2=256, OUT=64, 256 experts.
// Strategy: bucket rows by expert, then 1 workgroup per expert runs 16-row
// GEMM tiles through V_WMMA_F32_16X16X4_F32 (full f32, no downcast).
// Layers 1-2: each wave computes TWO output-column tiles per K-pass (dual
// accumulators -> 2 independent WMMA chains, A-fragments read from LDS once).
// Activations use native V_TANH_F32 when the builtin is available.
// ---------------------------------------------------------------------------

#define N_ROWS   16384
#define IN_DIM   64
#define H_DIM    256
#define OUT_DIM  64
#define N_EXP    256

typedef float v2f __attribute__((ext_vector_type(2)));
typedef float v8f __attribute__((ext_vector_type(8)));

#if __has_builtin(__builtin_amdgcn_tanhf)
__device__ __forceinline__ float fast_tanh(float v) { return __builtin_amdgcn_tanhf(v); }
#else
__device__ __forceinline__ float fast_tanh(float v) { return tanhf(v); }
#endif

// ------------------------- binning kernels ---------------------------------

__global__ __launch_bounds__(256) void jsmlp_init(int* counts, int* cursor) {
    int t = blockIdx.x * blockDim.x + threadIdx.x;
    if (t < N_EXP) { counts[t] = 0; cursor[t] = 0; }
}

__global__ __launch_bounds__(256) void jsmlp_hist(const int* __restrict__ ind,
                                                  int* __restrict__ counts) {
    int n = blockIdx.x * blockDim.x + threadIdx.x;
    if (n < N_ROWS) atomicAdd(&counts[ind[n]], 1);
}

// single block of 256 threads: exclusive scan of counts -> offsets[0..256]
__global__ __launch_bounds__(256) void jsmlp_scan(const int* __restrict__ counts,
                                                  int* __restrict__ offsets) {
    __shared__ int tmp[N_EXP];
    int t = threadIdx.x;
    tmp[t] = counts[t];
    __syncthreads();
    for (int d = 1; d < N_EXP; d <<= 1) {
        int v = (t >= d) ? tmp[t - d] : 0;
        __syncthreads();
        tmp[t] += v;
        __syncthreads();
    }
    offsets[t + 1] = tmp[t];          // inclusive -> offsets[t+1]
    if (t == 0) offsets[0] = 0;
}

__global__ __launch_bounds__(256) void jsmlp_scatter(const int* __restrict__ ind,
                                                     const int* __restrict__ offsets,
                                                     int* __restrict__ cursor,
                                                     int* __restrict__ rowidx) {
    int n = blockIdx.x * blockDim.x + threadIdx.x;
    if (n < N_ROWS) {
        int e = ind[n];
        int pos = atomicAdd(&cursor[e], 1);
        rowidx[offsets[e] + pos] = n;
    }
}

// ------------------------- expert MLP kernel -------------------------------
// grid = 256 blocks (one per expert), 256 threads (8 wave32).

__global__ __launch_bounds__(256) void jsmlp_expert(
    const float* __restrict__ x,
    const float* __restrict__ W1, const float* __restrict__ b1,
    const float* __restrict__ W2, const float* __restrict__ b2,
    const float* __restrict__ Wl, const float* __restrict__ bl,
    const int* __restrict__ offsets, const int* __restrict__ rowidx,
    float* __restrict__ out)
{
    __shared__ float xs[16][IN_DIM];   // 4 KB  : 16-row input tile
    __shared__ float h1[16][H_DIM];    // 16 KB : layer-1 activations
    __shared__ float h2[16][H_DIM];    // 16 KB : layer-2 activations
    __shared__ int   rows[16];

    const int e    = blockIdx.x;
    const int tid  = threadIdx.x;
    const int wave = tid >> 5;
    const int lane = tid & 31;
    const int hi   = lane >> 4;   // 0: lanes 0-15, 1: lanes 16-31
    const int lo   = lane & 15;   // M (A-frag) / N (B,C,D frags)

    const int beg = offsets[e];
    const int cnt = offsets[e + 1] - beg;
    if (cnt == 0) return;

    const float* W1e = W1 + (size_t)e * H_DIM * IN_DIM;
    const float* b1e = b1 + (size_t)e * H_DIM;
    const float* W2e = W2 + (size_t)e * H_DIM * H_DIM;
    const float* b2e = b2 + (size_t)e * H_DIM;
    const float* Wle = Wl + (size_t)e * OUT_DIM * H_DIM;
    const float* ble = bl + (size_t)e * OUT_DIM;

    // the two output-column tiles this wave owns in a 256-wide layer
    const int nc0 = wave * 16 + lo;          // tiles 0..7
    const int nc1 = (wave + 8) * 16 + lo;    // tiles 8..15

    for (int t0 = 0; t0 < cnt; t0 += 16) {
        __syncthreads();                       // prev tile fully consumed
        if (tid < 16)
            rows[tid] = (t0 + tid < cnt) ? rowidx[beg + t0 + tid] : -1;
        __syncthreads();

        // cooperative x-tile load: 16*64 floats, 4 per thread (float4)
        {
            int idx = tid * 4;                 // [0,1024)
            int r   = idx >> 6;
            int c   = idx & 63;
            int g   = rows[r];
            float4 v = {0.f, 0.f, 0.f, 0.f};
            if (g >= 0) v = *(const float4*)(x + (size_t)g * IN_DIM + c);
            *(float4*)&xs[r][c] = v;
        }
        __syncthreads();

        // ---- layer 1: h1 = tanh(xs(16x64) @ W1^T(64x256) + b1) ----
        {
            v8f acc0 = {}, acc1 = {};
            #pragma unroll
            for (int kb = 0; kb < IN_DIM; kb += 4) {
                int k = kb + 2 * hi;           // K for this lane's 2 VGPRs
                v2f a  = *(const v2f*)&xs[lo][k];
                v2f b0 = *(const v2f*)(W1e + (size_t)nc0 * IN_DIM + k);
                v2f b1f= *(const v2f*)(W1e + (size_t)nc1 * IN_DIM + k);
                acc0 = __builtin_amdgcn_wmma_f32_16x16x4_f32(
                         false, a, false, b0,  (short)0, acc0, false, false);
                acc1 = __builtin_amdgcn_wmma_f32_16x16x4_f32(
                         false, a, false, b1f, (short)0, acc1, false, false);
            }
            float bias0 = b1e[nc0];
            float bias1 = b1e[nc1];
            #pragma unroll
            for (int r = 0; r < 8; ++r) {
                h1[r + 8 * hi][nc0] = fast_tanh(acc0[r] + bias0);
                h1[r + 8 * hi][nc1] = fast_tanh(acc1[r] + bias1);
            }
        }
        __syncthreads();

        // ---- layer 2: h2 = tanh(h1(16x256) @ W2^T(256x256) + b2) ----
        {
            v8f acc0 = {}, acc1 = {};
            #pragma unroll 8
            for (int kb = 0; kb < H_DIM; kb += 4) {
                int k = kb + 2 * hi;
                v2f a  = *(const v2f*)&h1[lo][k];
                v2f b0 = *(const v2f*)(W2e + (size_t)nc0 * H_DIM + k);
                v2f b1f= *(const v2f*)(W2e + (size_t)nc1 * H_DIM + k);
                acc0 = __builtin_amdgcn_wmma_f32_16x16x4_f32(
                         false, a, false, b0,  (short)0, acc0, false, false);
                acc1 = __builtin_amdgcn_wmma_f32_16x16x4_f32(
                         false, a, false, b1f, (short)0, acc1, false, false);
            }
            float bias0 = b2e[nc0];
            float bias1 = b2e[nc1];
            #pragma unroll
            for (int r = 0; r < 8; ++r) {
                h2[r + 8 * hi][nc0] = fast_tanh(acc0[r] + bias0);
                h2[r + 8 * hi][nc1] = fast_tanh(acc1[r] + bias1);
            }
        }
        __syncthreads();

        // ---- layer 3: out = h2(16x256) @ Wl^T(256x64) + bl ----
        if (wave < OUT_DIM / 16) {
            const int ncol = wave * 16 + lo;
            v8f acc = {};
            #pragma unroll 8
            for (int kb = 0; kb < H_DIM; kb += 4) {
                int k = kb + 2 * hi;
                v2f a = *(const v2f*)&h2[lo][k];
                v2f b = *(const v2f*)(Wle + (size_t)ncol * H_DIM + k);
                acc = __builtin_amdgcn_wmma_f32_16x16x4_f32(
                        false, a, false, b, (short)0, acc, false, false);
            }
            float bias = ble[ncol];
            #pragma unroll
            for (int r = 0; r < 8; ++r) {
                int g = rows[r + 8 * hi];
                if (g >= 0)
                    out[(size_t)g * OUT_DIM + ncol] = acc[r] + bias;
            }
        }
    }
}

// ------------------------------- launcher ----------------------------------

extern "C" void kernel_launch(void* const* d_in, const int* in_sizes, int n_in,
                              void* d_out, int out_size, void* d_ws, size_t ws_size,
                              hipStream_t stream) {
    const float* x   = (const float*)d_in[0];
    const int*   ind = (const int*)  d_in[1];
    const float* W1  = (const float*)d_in[2];
    const float* b1  = (const float*)d_in[3];
    const float* W2  = (const float*)d_in[4];
    const float* b2  = (const float*)d_in[5];
    const float* Wl  = (const float*)d_in[6];
    const float* bl  = (const float*)d_in[7];
    float* out = (float*)d_out;

    int* ws      = (int*)d_ws;
    int* counts  = ws;            // 256
    int* cursor  = ws + 256;      // 256
    int* offsets = ws + 512;      // 257
    int* rowidx  = ws + 1024;     // 16384

    jsmlp_init   <<<1, 256, 0, stream>>>(counts, cursor);
    jsmlp_hist   <<<N_ROWS / 256, 256, 0, stream>>>(ind, counts);
    jsmlp_scan   <<<1, 256, 0, stream>>>(counts, offsets);
    jsmlp_scatter<<<N_ROWS / 256, 256, 0, stream>>>(ind, offsets, cursor, rowidx);
    jsmlp_expert <<<N_EXP, 256, 0, stream>>>(x, W1, b1, W2, b2, Wl, bl,
                                             offsets, rowidx, out);
}